// FAPELoss_42099269435584
// MI455X (gfx1250) — compile-verified
//
#include <hip/hip_runtime.h>
#include <hip/hip_bf16.h>

#define NATOMS   4096
#define NFRAMES  4094
#define FPAD     4096
#define NPADF    (FPAD - NFRAMES)   // 2 zero-padded frames
#define KDIM     20
#define CLAMP_D  10.0f
#define EPSF     1e-8f

typedef float v2f __attribute__((ext_vector_type(2)));
typedef float v8f __attribute__((ext_vector_type(8)));

// ---------------------------------------------------------------------------
// Kernel 1: per-atom features  Afeat[k][n], k = 0..19, feature-major.
//   k0..2  = p, k3..5 = t, k6..14 = p_i * t_j, k15 = 1, k16 = |p|^2+|t|^2
// ---------------------------------------------------------------------------
__global__ void fape_atom_feats(const float* __restrict__ pred,
                                const float* __restrict__ tru,
                                float* __restrict__ af) {
  int n = blockIdx.x * blockDim.x + threadIdx.x;
  if (n >= NATOMS) return;
  float p[3] = {pred[3*n], pred[3*n+1], pred[3*n+2]};
  float t[3] = {tru[3*n],  tru[3*n+1],  tru[3*n+2]};
  af[0*NATOMS+n] = p[0]; af[1*NATOMS+n] = p[1]; af[2*NATOMS+n] = p[2];
  af[3*NATOMS+n] = t[0]; af[4*NATOMS+n] = t[1]; af[5*NATOMS+n] = t[2];
  #pragma unroll
  for (int i = 0; i < 3; ++i)
    #pragma unroll
    for (int j = 0; j < 3; ++j)
      af[(6 + i*3 + j)*NATOMS + n] = p[i] * t[j];
  af[15*NATOMS+n] = 1.0f;
  af[16*NATOMS+n] = p[0]*p[0]+p[1]*p[1]+p[2]*p[2] + t[0]*t[0]+t[1]*t[1]+t[2]*t[2];
  af[17*NATOMS+n] = 0.0f; af[18*NATOMS+n] = 0.0f; af[19*NATOMS+n] = 0.0f;
}

// ---------------------------------------------------------------------------
// Kernel 2: per-frame features Bfeat[k][f], feature-major, padded to FPAD
// with zeros. Zero rows make padded frames' d^2 exactly 0.0, so their
// contribution is the exact constant min(v_sqrt(EPS),10), corrected once in
// the final reduction (no per-element masking in the hot loop).
// ---------------------------------------------------------------------------
__device__ __forceinline__ void build_frame(const float* __restrict__ c, int f,
                                            float o[3], float R[3][3]) {
  float a0 = c[3*f+0],     a1 = c[3*f+1],     a2 = c[3*f+2];
  float b0 = c[3*(f+1)+0], b1 = c[3*(f+1)+1], b2 = c[3*(f+1)+2];
  float d0 = c[3*(f+2)+0], d1 = c[3*(f+2)+1], d2 = c[3*(f+2)+2];
  o[0] = b0; o[1] = b1; o[2] = b2;
  float e1x = d0-b0, e1y = d1-b1, e1z = d2-b2;
  float n1 = sqrtf(e1x*e1x + e1y*e1y + e1z*e1z) + EPSF;
  e1x /= n1; e1y /= n1; e1z /= n1;
  float e2x = a0-b0, e2y = a1-b1, e2z = a2-b2;
  float dp = e2x*e1x + e2y*e1y + e2z*e1z;
  e2x -= dp*e1x; e2y -= dp*e1y; e2z -= dp*e1z;
  float n2 = sqrtf(e2x*e2x + e2y*e2y + e2z*e2z) + EPSF;
  e2x /= n2; e2y /= n2; e2z /= n2;
  float e3x = e1y*e2z - e1z*e2y;
  float e3y = e1z*e2x - e1x*e2z;
  float e3z = e1x*e2y - e1y*e2x;
  R[0][0]=e1x; R[0][1]=e1y; R[0][2]=e1z;
  R[1][0]=e2x; R[1][1]=e2y; R[1][2]=e2z;
  R[2][0]=e3x; R[2][1]=e3y; R[2][2]=e3z;
}

__global__ void fape_frame_feats(const float* __restrict__ pred,
                                 const float* __restrict__ tru,
                                 float* __restrict__ ff) {
  int f = blockIdx.x * blockDim.x + threadIdx.x;
  if (f >= FPAD) return;
  if (f >= NFRAMES) {
    #pragma unroll
    for (int k = 0; k < KDIM; ++k) ff[k*FPAD + f] = 0.0f;
    return;
  }
  float op[3], ot[3], Rp[3][3], Rt[3][3];
  build_frame(pred, f, op, Rp);
  build_frame(tru,  f, ot, Rt);
  // M = Rp^T * Rt  (3x3):  M[i][j] = sum_r Rp[r][i] * Rt[r][j]
  float M[3][3];
  #pragma unroll
  for (int i = 0; i < 3; ++i)
    #pragma unroll
    for (int j = 0; j < 3; ++j)
      M[i][j] = Rp[0][i]*Rt[0][j] + Rp[1][i]*Rt[1][j] + Rp[2][i]*Rt[2][j];
  float Mot[3], MTop[3];
  #pragma unroll
  for (int i = 0; i < 3; ++i) {
    Mot[i]  = M[i][0]*ot[0] + M[i][1]*ot[1] + M[i][2]*ot[2];
    MTop[i] = M[0][i]*op[0] + M[1][i]*op[1] + M[2][i]*op[2];
  }
  ff[0*FPAD+f] = 2.0f*(Mot[0]  - op[0]);
  ff[1*FPAD+f] = 2.0f*(Mot[1]  - op[1]);
  ff[2*FPAD+f] = 2.0f*(Mot[2]  - op[2]);
  ff[3*FPAD+f] = 2.0f*(MTop[0] - ot[0]);
  ff[4*FPAD+f] = 2.0f*(MTop[1] - ot[1]);
  ff[5*FPAD+f] = 2.0f*(MTop[2] - ot[2]);
  #pragma unroll
  for (int i = 0; i < 3; ++i)
    #pragma unroll
    for (int j = 0; j < 3; ++j)
      ff[(6 + i*3 + j)*FPAD + f] = -2.0f * M[i][j];
  float opn = op[0]*op[0]+op[1]*op[1]+op[2]*op[2];
  float otn = ot[0]*ot[0]+ot[1]*ot[1]+ot[2]*ot[2];
  float opMot = op[0]*Mot[0]+op[1]*Mot[1]+op[2]*Mot[2];
  ff[15*FPAD+f] = opn + otn - 2.0f*opMot;
  ff[16*FPAD+f] = 1.0f;
  ff[17*FPAD+f] = 0.0f; ff[18*FPAD+f] = 0.0f; ff[19*FPAD+f] = 0.0f;
}

// ---------------------------------------------------------------------------
// Kernel 3: main WMMA sweep. Each wave32 owns one 16-atom tile (B operand,
// kept in registers) and iterates 16 frame tiles of 16 frames (A operand).
// D = A(16 frames x K) * B(K x 16 atoms) accumulated over K=20 via five
// chained V_WMMA_F32_16X16X4_F32. Epilogue: acc += min(v_sqrt(d2+eps), 10),
// unconditionally (padded frames corrected analytically in kernel 4).
// 512 blocks x 8 waves = 4096 waves = 256 atom tiles x 16 frame chunks.
// ---------------------------------------------------------------------------
__global__ __launch_bounds__(256) void fape_main(const float* __restrict__ af,
                                                 const float* __restrict__ ff,
                                                 float* __restrict__ partials) {
  int tid    = threadIdx.x;
  int lane   = tid & 31;
  int waveId = blockIdx.x * 8 + (tid >> 5);
  int atile  = waveId & 255;   // 256 atom tiles
  int chunk  = waveId >> 8;    // 16 frame chunks of 16 tiles each
  int l16    = lane & 15;
  int hi     = lane >> 4;      // 0: K 0..1 / M 0..7, 1: K 2..3 / M 8..15
  int kofs   = hi * 2;

  // B operand: atom features for this tile, resident for the whole sweep.
  int n = atile * 16 + l16;
  v2f b[5];
  #pragma unroll
  for (int kc = 0; kc < 5; ++kc) {
    b[kc].x = af[(kc*4 + kofs + 0) * NATOMS + n];
    b[kc].y = af[(kc*4 + kofs + 1) * NATOMS + n];
  }

  float acc = 0.0f;
  for (int i = 0; i < 16; ++i) {
    int ftile = chunk * 16 + i;
    int fl    = ftile * 16 + l16;
    v2f a[5];
    #pragma unroll
    for (int kc = 0; kc < 5; ++kc) {
      a[kc].x = ff[(kc*4 + kofs + 0) * FPAD + fl];
      a[kc].y = ff[(kc*4 + kofs + 1) * FPAD + fl];
    }
    v8f c = {};
    c = __builtin_amdgcn_wmma_f32_16x16x4_f32(false, a[0], false, b[0], (short)0, c, false, false);
    c = __builtin_amdgcn_wmma_f32_16x16x4_f32(false, a[1], false, b[1], (short)0, c, false, false);
    c = __builtin_amdgcn_wmma_f32_16x16x4_f32(false, a[2], false, b[2], (short)0, c, false, false);
    c = __builtin_amdgcn_wmma_f32_16x16x4_f32(false, a[3], false, b[3], (short)0, c, false, false);
    c = __builtin_amdgcn_wmma_f32_16x16x4_f32(false, a[4], false, b[4], (short)0, c, false, false);
    // Epilogue: no masking, no index math. EXEC stays all-ones throughout.
    #pragma unroll
    for (int r = 0; r < 8; ++r) {
      float d2 = fmaxf(c[r], 0.0f);                    // guard rounding underflow
      acc += fminf(__builtin_amdgcn_sqrtf(d2 + EPSF), CLAMP_D); // raw v_sqrt_f32
    }
  }

  __shared__ float red[256];
  red[tid] = acc;
  __syncthreads();
  #pragma unroll
  for (int s = 128; s > 0; s >>= 1) {
    if (tid < s) red[tid] += red[tid + s];
    __syncthreads();
  }
  if (tid == 0) partials[blockIdx.x] = red[0];
}

// ---------------------------------------------------------------------------
// Kernel 4: deterministic final reduction of the 512 block partials.
// Subtract the exact contribution of the 2 zero-padded frames (d^2 == 0.0
// exactly, so each padded element added exactly min(v_sqrt(EPS),10) —
// computed here with the same instruction for a bit-matching correction).
// Mean over F*N = 4094*4096 (exactly representable in f32), then /10.
// ---------------------------------------------------------------------------
__global__ void fape_reduce(const float* __restrict__ partials,
                            float* __restrict__ out) {
  __shared__ float red[512];
  int tid = threadIdx.x;
  red[tid] = partials[tid];
  __syncthreads();
  #pragma unroll
  for (int s = 256; s > 0; s >>= 1) {
    if (tid < s) red[tid] += red[tid + s];
    __syncthreads();
  }
  if (tid == 0) {
    float pad_elem = fminf(__builtin_amdgcn_sqrtf(EPSF), CLAMP_D);
    float total = red[0] - (float)(NPADF * NATOMS) * pad_elem;
    out[0] = total / ((float)NFRAMES * (float)NATOMS) / 10.0f;
  }
}

extern "C" void kernel_launch(void* const* d_in, const int* in_sizes, int n_in,
                              void* d_out, int out_size, void* d_ws, size_t ws_size,
                              hipStream_t stream) {
  const float* pred = (const float*)d_in[0];
  const float* tru  = (const float*)d_in[1];
  // Workspace layout (floats): [atom feats 20*4096][frame feats 20*4096][512 partials]
  float* af       = (float*)d_ws;
  float* ff       = af + KDIM * NATOMS;
  float* partials = ff + KDIM * FPAD;

  fape_atom_feats <<<NATOMS/256, 256, 0, stream>>>(pred, tru, af);
  fape_frame_feats<<<FPAD/256,   256, 0, stream>>>(pred, tru, ff);
  fape_main       <<<512,        256, 0, stream>>>(af, ff, partials);
  fape_reduce     <<<1,          512, 0, stream>>>(partials, (float*)d_out);
}